// SoftSubSampler_3487513445108
// MI455X (gfx1250) — compile-verified
//
#include <hip/hip_runtime.h>
#include <cstdint>
#include <cstddef>

namespace {

constexpr int   kB   = 64;
constexpr int   kN   = 500000;
constexpr int   kK   = 10;
constexpr float kInvT = 2.0f;                       // 1 / T, T = 0.5
constexpr float kEps  = 1.1920928955078125e-07f;    // fp32 eps (torch clamp_probs)

constexpr int TPB_TOPK = 512;
constexpr int TPB      = 1024;                      // 32 wave32s per WGP
constexpr int N4       = kN / 4;                    // 125000 float4s per row
constexpr int NCHUNK   = (N4 + TPB - 1) / TPB;      // 123 chunks (last partial: 72 lanes)
constexpr int DEPTH    = 4;                         // async pipeline depth

__device__ __forceinline__ float clampp(float p) {
  return fminf(fmaxf(p, kEps), 1.0f - kEps);
}

// branchless online softmax accumulator: running max m, running sum s
__device__ __forceinline__ void online_ms(float& m, float& s, float x) {
  float M = fmaxf(m, x);
  s = s * __expf(m - M) + __expf(x - M);
  m = M;
}

__device__ __forceinline__ void merge_ms(float& m, float& s, float m2, float s2) {
  float M = fmaxf(m, m2);
  s = s * __expf(m - M) + s2 * __expf(m2 - M);
  m = M;
}

// CDNA5 async global -> LDS (128-bit per lane), tracked by ASYNCcnt.
__device__ __forceinline__ void async_ld_f4(const float4* gsrc, const float4* lds_slot) {
  // generic pointer to LDS: low 32 bits are the workgroup-relative LDS byte addr
  uint32_t lds = (uint32_t)(uintptr_t)lds_slot;
  asm volatile("global_load_async_to_lds_b128 %0, %1, off"
               :: "v"(lds), "v"(gsrc)
               : "memory");
}

template <int N>
__device__ __forceinline__ void wait_async() {
  asm volatile("s_wait_asynccnt %0" :: "n"(N) : "memory");
}
__device__ __forceinline__ void wait_store_0() {
  asm volatile("s_wait_storecnt 0" ::: "memory");
}

// ---------------------------------------------------------------------------
// Kernel 1: per-row K-th largest threshold + dsamples = (logits >= thr)
// ---------------------------------------------------------------------------
__global__ __launch_bounds__(TPB_TOPK) void topk_kernel(
    const float* __restrict__ logits, float* __restrict__ dsamples) {
  __shared__ float lists[TPB_TOPK][kK];
  __shared__ float thr_s;

  const int b = blockIdx.x;
  const int t = threadIdx.x;
  const float*  row  = logits + (size_t)b * kN;
  const float4* row4 = (const float4*)row;

  // Per-thread sorted-descending top-10, fully unrolled (static reg indexing).
  float r[kK];
#pragma unroll
  for (int i = 0; i < kK; ++i) r[i] = -__builtin_inff();

#pragma unroll 2
  for (int i = t; i < N4; i += TPB_TOPK) {
    float4 v4 = row4[i];
    float vals[4] = {v4.x, v4.y, v4.z, v4.w};
#pragma unroll
    for (int c = 0; c < 4; ++c) {
      float v = vals[c];
      if (v > r[kK - 1]) {
#pragma unroll
        for (int j = 0; j < kK; ++j) {   // bubble-insert, keeps r sorted desc
          float old = r[j];
          bool  gt  = v > old;
          r[j] = gt ? v : old;
          v    = gt ? old : v;
        }
      }
    }
  }

#pragma unroll
  for (int i = 0; i < kK; ++i) lists[t][i] = r[i];
  __syncthreads();

  // LDS tree merge of sorted-descending 10-lists (two-pointer, keep top 10).
  for (int stride = TPB_TOPK / 2; stride >= 1; stride >>= 1) {
    if (t < stride) {
      float out[kK];
      int ia = 0, ib = 0;
#pragma unroll
      for (int k = 0; k < kK; ++k) {
        float a = lists[t][ia];
        float c = lists[t + stride][ib];
        if (a >= c) { out[k] = a; ++ia; } else { out[k] = c; ++ib; }
      }
#pragma unroll
      for (int k = 0; k < kK; ++k) lists[t][k] = out[k];
    }
    __syncthreads();
  }

  if (t == 0) thr_s = lists[0][kK - 1];
  __syncthreads();
  const float thr = thr_s;

  float4* d4 = (float4*)(dsamples + (size_t)b * kN);
#pragma unroll 4
  for (int i = t; i < N4; i += TPB_TOPK) {
    float4 v = row4[i];
    float4 o;
    o.x = (v.x >= thr) ? 1.0f : 0.0f;
    o.y = (v.y >= thr) ? 1.0f : 0.0f;
    o.z = (v.z >= thr) ? 1.0f : 0.0f;
    o.w = (v.w >= thr) ? 1.0f : 0.0f;
    d4[i] = o;
  }
}

// ---------------------------------------------------------------------------
// Kernel 2: iterative Gumbel-softmax continuous top-k. One block per row.
// w (state, 128 MB total -> L2-resident) lives in d_ws.
// ---------------------------------------------------------------------------
__global__ __launch_bounds__(TPB) void cont_topk_kernel(
    const float* __restrict__ logits, const float* __restrict__ u,
    float* __restrict__ w, float* __restrict__ cs) {
  __shared__ float4 stage[DEPTH][TPB];  // async pipeline stage (64 KB)
  __shared__ float  red_m[TPB / 32];
  __shared__ float  red_s[TPB / 32];
  __shared__ float  MS[2];

  const int b = blockIdx.x;
  const int t = threadIdx.x;
  const size_t roff = (size_t)b * kN;
  const float4* l4 = (const float4*)(logits + roff);
  const float4* u4 = (const float4*)(u + roff);
  float4* w4 = (float4*)(w + roff);
  float4* c4 = (float4*)(cs + roff);

  for (int it = 0; it < kK; ++it) {
    // ---- pass A: online row max/sumexp of x = w * (1/T) ------------------
    float m = -__builtin_inff();
    float s = 0.0f;

    if (it == 0) {
      // fuse Gumbel-noise construction: w = logits - log(-log(clamp(u)))
#pragma unroll 2
      for (int i = t; i < N4; i += TPB) {
        float4 lv = l4[i];
        float4 uv = u4[i];
        float4 wv;
        wv.x = lv.x - __logf(-__logf(clampp(uv.x)));
        wv.y = lv.y - __logf(-__logf(clampp(uv.y)));
        wv.z = lv.z - __logf(-__logf(clampp(uv.z)));
        wv.w = lv.w - __logf(-__logf(clampp(uv.w)));
        w4[i] = wv;
        online_ms(m, s, wv.x * kInvT);
        online_ms(m, s, wv.y * kInvT);
        online_ms(m, s, wv.z * kInvT);
        online_ms(m, s, wv.w * kInvT);
      }
    } else {
      // order async reads after last iteration's w stores (async ops are
      // unordered vs regular VMEM stores per ISA ch.10.7)
      wait_store_0();

      // DEPTH-deep async global->LDS pipeline. Addresses are lane-clamped on
      // the final partial chunk so every wave issues every chunk and the
      // per-wave ASYNCcnt bookkeeping stays uniform; consumption is guarded.
#pragma unroll
      for (int c = 0; c < DEPTH; ++c) {
        int idx = min(c * TPB + t, N4 - 1);
        async_ld_f4(w4 + idx, &stage[c][t]);
      }
      for (int c = 0; c < NCHUNK; ++c) {
        const int rem = NCHUNK - 1 - c;          // chunks still in flight after c
        if (rem >= DEPTH - 1)      wait_async<DEPTH - 1>();
        else if (rem == 2)         wait_async<2>();
        else if (rem == 1)         wait_async<1>();
        else                       wait_async<0>();

        int idx = c * TPB + t;
        if (idx < N4) {
          float4 wv = stage[c & (DEPTH - 1)][t];
          online_ms(m, s, wv.x * kInvT);
          online_ms(m, s, wv.y * kInvT);
          online_ms(m, s, wv.z * kInvT);
          online_ms(m, s, wv.w * kInvT);
        }
        if (c + DEPTH < NCHUNK) {                // refill the slot just drained
          int nidx = min((c + DEPTH) * TPB + t, N4 - 1);
          async_ld_f4(w4 + nidx, &stage[c & (DEPTH - 1)][t]);
        }
      }
    }

    // ---- reduce (m, s) across 32 lanes, then across 32 waves --------------
#pragma unroll
    for (int off = 16; off >= 1; off >>= 1) {
      float m2 = __shfl_xor(m, off, 32);
      float s2 = __shfl_xor(s, off, 32);
      merge_ms(m, s, m2, s2);
    }
    if ((t & 31) == 0) { red_m[t >> 5] = m; red_s[t >> 5] = s; }
    __syncthreads();
    if (t < 32) {
      m = red_m[t];
      s = red_s[t];
#pragma unroll
      for (int off = 16; off >= 1; off >>= 1) {
        float m2 = __shfl_xor(m, off, 32);
        float s2 = __shfl_xor(s, off, 32);
        merge_ms(m, s, m2, s2);
      }
      if (t == 0) { MS[0] = m; MS[1] = s; }
    }
    __syncthreads();
    const float M    = MS[0];
    const float invS = 1.0f / MS[1];

    // ---- pass B: p = softmax, acc += p, w += log(clamp(1 - p)) ------------
#pragma unroll 4
    for (int i = t; i < N4; i += TPB) {
      float4 wv = w4[i];
      float4 pv;
      pv.x = __expf(wv.x * kInvT - M) * invS;
      pv.y = __expf(wv.y * kInvT - M) * invS;
      pv.z = __expf(wv.z * kInvT - M) * invS;
      pv.w = __expf(wv.w * kInvT - M) * invS;

      float4 av;
      if (it == 0) {
        av = pv;                         // first write also initializes d_out
      } else {
        float4 prev = c4[i];
        av.x = prev.x + pv.x; av.y = prev.y + pv.y;
        av.z = prev.z + pv.z; av.w = prev.w + pv.w;
      }
      c4[i] = av;

      if (it < kK - 1) {
        wv.x += __logf(clampp(1.0f - pv.x));
        wv.y += __logf(clampp(1.0f - pv.y));
        wv.z += __logf(clampp(1.0f - pv.z));
        wv.w += __logf(clampp(1.0f - pv.w));
        w4[i] = wv;
      }
    }
    // next iteration's first barrier orders LDS reuse; w elements are
    // single-owner per thread, so no extra block barrier is needed here.
  }
}

}  // namespace

extern "C" void kernel_launch(void* const* d_in, const int* in_sizes, int n_in,
                              void* d_out, int out_size, void* d_ws, size_t ws_size,
                              hipStream_t stream) {
  (void)in_sizes; (void)n_in; (void)out_size; (void)ws_size;
  const float* logits = (const float*)d_in[0];
  const float* u      = (const float*)d_in[1];
  float* dsamples = (float*)d_out;                       // [B*N]
  float* csamples = (float*)d_out + (size_t)kB * kN;     // [B*N]
  float* w        = (float*)d_ws;                        // [B*N] fp32 state

  topk_kernel<<<kB, TPB_TOPK, 0, stream>>>(logits, dsamples);
  cont_topk_kernel<<<kB, TPB, 0, stream>>>(logits, u, w, csamples);
}